// BKT_4698694221978
// MI455X (gfx1250) — compile-verified
//
#include <hip/hip_runtime.h>
#include <cstdint>

// BKT forward scan on MI455X (gfx1250).
// Memory-bound streaming scan: ~420 MB traffic -> ~18us floor @ 23.3 TB/s.
// Strategy: per-thread 4-column state, depth-8 async global->LDS pipeline
// (global_load_async_to_lds_b128 + s_wait_asynccnt), wave-private LDS slots
// (no barriers), rcp+Newton for the per-step division, nontemporal stores.
// Depth 8 -> ~8 MB in flight chip-wide (2048 waves x 8 x 512 B), enough to
// cover HBM latency at 23.3 TB/s.

#define NA   200
#define BATCH 262144
#define TPB  256
#define CPT  4      // columns per thread
#define DEPTH 8     // async pipeline depth (power of 2)

typedef __attribute__((ext_vector_type(4))) float v4f;  // native vector for NT store

__device__ __forceinline__ float fast_rcp(float d) {
    // v_rcp_f32 (~1 ulp) + one Newton-Raphson step -> ~0.5 ulp
    float r = __builtin_amdgcn_rcpf(d);
    r = r * (2.0f - d * r);
    return r;
}

__device__ __forceinline__ float sigmoidf_precise(float v) {
    return 1.0f / (1.0f + expf(-v));
}

__global__ __launch_bounds__(TPB) void bkt_scan_kernel(
    const int* __restrict__ x,
    const float* __restrict__ pL0, const float* __restrict__ pT,
    const float* __restrict__ pF,  const float* __restrict__ pG,
    const float* __restrict__ pS,
    float* __restrict__ y)
{
    // DEPTH buffers x TPB lanes x 16B = 32 KB LDS per block.
    __shared__ uint4 tile[DEPTH][TPB];

    const int tid  = threadIdx.x;
    const int gtid = blockIdx.x * TPB + tid;
    const size_t gcol = (size_t)gtid * CPT;

    // Scalar learned params: uniform -> compiler lowers sigmoid to SALU float ops.
    const float trans  = sigmoidf_precise(pT[0]);
    const float forget = sigmoidf_precise(pF[0]);
    const float guess  = sigmoidf_precise(pG[0]);
    const float slip   = sigmoidf_precise(pS[0]);
    const float l0     = sigmoidf_precise(pL0[0]);

    const float oms  = 1.0f - slip;           // 1 - slip
    const float omsg = oms - guess;           // (1-slip) - guess
    const float kk   = 1.0f - forget - trans; // blend coefficient for state update

    // AS3 byte offset of this lane's slot (low 32 bits of the flat LDS address).
    unsigned ldsbase = (unsigned)(size_t)(&tile[0][0]) + (unsigned)tid * 16u;

    float learn[CPT];
#pragma unroll
    for (int j = 0; j < CPT; ++j) learn[j] = l0;

    // Issue one async 16B copy global->LDS for time step t into buffer t%DEPTH.
    auto issue = [&](int t) {
        unsigned la = ldsbase + (unsigned)(t & (DEPTH - 1)) * (TPB * 16u);
        unsigned long long ga =
            (unsigned long long)(uintptr_t)(x + (size_t)t * BATCH + gcol);
        asm volatile("global_load_async_to_lds_b128 %0, %1, off"
                     :: "v"(la), "v"(ga) : "memory");
    };

    // Prologue: fill pipeline (steps 0 .. DEPTH-2).
#pragma unroll
    for (int p = 0; p < DEPTH - 1; ++p) issue(p);

    for (int t = 0; t < NA; ++t) {
        if (t + DEPTH - 1 < NA) {
            issue(t + DEPTH - 1);
            // <=7 outstanding -> the oldest (step t) has landed in LDS.
            asm volatile("s_wait_asynccnt 7" ::: "memory");
        } else {
            // Tail: drain fully (last DEPTH-1 iterations only).
            asm volatile("s_wait_asynccnt 0" ::: "memory");
        }

        const uint4 xv = tile[t & (DEPTH - 1)][tid];   // ds_load_b128
        const unsigned xs[CPT] = {xv.x, xv.y, xv.z, xv.w};

        float out[CPT];
#pragma unroll
        for (int j = 0; j < CPT; ++j) {
            const bool  b       = (xs[j] != 0u);
            const float lj      = learn[j];
            // correct = lj*(1-slip) + (1-lj)*guess = lj*((1-slip)-guess) + guess
            const float correct = fmaf(lj, omsg, guess);
            out[j] = correct;
            // x==1: cond = lj*(1-slip)/correct ; x==0: cond = lj*slip/(1-correct)
            const float numer = lj * (b ? oms : slip);
            const float denom = b ? correct : (1.0f - correct);
            const float cond  = numer * fast_rcp(denom);
            // new_learn = cond*(1-forget) + (1-cond)*trans = cond*(1-f-tau) + tau
            learn[j] = fmaf(cond, kk, trans);
        }

        v4f o = {out[0], out[1], out[2], out[3]};
        // Streaming output: write-once, never re-read on device -> NT store.
        __builtin_nontemporal_store(o, (v4f*)(y + (size_t)t * BATCH + gcol));
    }
}

extern "C" void kernel_launch(void* const* d_in, const int* in_sizes, int n_in,
                              void* d_out, int out_size, void* d_ws, size_t ws_size,
                              hipStream_t stream) {
    const int*   x  = (const int*)d_in[0];
    const float* L0 = (const float*)d_in[1];
    const float* T  = (const float*)d_in[2];
    const float* F  = (const float*)d_in[3];
    const float* G  = (const float*)d_in[4];
    const float* S  = (const float*)d_in[5];
    float* y = (float*)d_out;

    dim3 grid(BATCH / (TPB * CPT));   // 262144 / 1024 = 256 blocks
    dim3 block(TPB);
    bkt_scan_kernel<<<grid, block, 0, stream>>>(x, L0, T, F, G, S, y);
}